// GPASS_90769838833732
// MI455X (gfx1250) — compile-verified
//
#include <hip/hip_runtime.h>
#include <hip/hip_bf16.h>

#define GAT_F 64
#define GAT_H 4
#define GAT_L 4
#define NEG_SLOPE 0.2f

typedef __attribute__((ext_vector_type(2))) float v2f;
typedef __attribute__((ext_vector_type(8))) float v8f;

// ---------------- init: zero rst/denom, set segment-max keys to enc(-inf) ----------------
__global__ void gat_init(float* __restrict__ rst, float* __restrict__ denom,
                         unsigned* __restrict__ keys, int nrst, int nnh) {
    int i = blockIdx.x * blockDim.x + threadIdx.x;
    int stride = gridDim.x * blockDim.x;
    for (int t = i; t < nrst; t += stride) rst[t] = 0.0f;
    for (int t = i; t < nnh; t += stride) { denom[t] = 0.0f; keys[t] = 0x007FFFFFu; } // enc(-inf)
}

// ---------------- GEMM: ft[N,256] = h[N,64] @ W[64,256] via fp32 WMMA ----------------
// One wave per 16x16 output tile, K=64 in 16 steps of V_WMMA_F32_16X16X4_F32.
__global__ __launch_bounds__(128)
void gat_gemm_wmma(const float* __restrict__ h, const float* __restrict__ W,
                   float* __restrict__ ft, int nNodes) {
    const int wave  = blockIdx.x * (blockDim.x >> 5) + (threadIdx.x >> 5);
    const int lane  = threadIdx.x & 31;
    const int mtile = wave >> 4;       // 16 n-tiles (256/16) per m-tile
    const int ntile = wave & 15;
    const int m0 = mtile * 16;
    if (m0 >= nNodes) return;          // wave-uniform; grid is exact for N=50000

    const int l15  = lane & 15;
    const int kb   = (lane >> 4) << 1; // 0 or 2
    const int mrow = m0 + l15;
    const int ncol = ntile * 16 + l15;
    const float* hrow = h + (size_t)mrow * GAT_F;

    v8f acc = {};
#pragma unroll
    for (int k = 0; k < GAT_F; k += 4) {
        v2f a = { hrow[k + kb], hrow[k + kb + 1] };
        v2f b = { W[(size_t)(k + kb) * 256 + ncol], W[(size_t)(k + kb + 1) * 256 + ncol] };
        acc = __builtin_amdgcn_wmma_f32_16x16x4_f32(false, a, false, b, (short)0, acc,
                                                    false, false);
    }
    const int rbase = m0 + ((lane >> 4) << 3);
#pragma unroll
    for (int r = 0; r < 8; ++r)
        ft[(size_t)(rbase + r) * 256 + ncol] = acc[r];
}

// ---------------- per (node,head) attention scores el/er ----------------
__global__ void gat_attn_scores(const float* __restrict__ ft, const float* __restrict__ al,
                                const float* __restrict__ ar, float* __restrict__ el,
                                float* __restrict__ er, int nnh) {
    int i = blockIdx.x * blockDim.x + threadIdx.x;
    if (i >= nnh) return;
    int hh = i & (GAT_H - 1);
    const float* f = ft + (size_t)i * GAT_F;   // ft[n][h][:] == flat (n*H+h)*F
    const float* a = al + hh * GAT_F;
    const float* b = ar + hh * GAT_F;
    float sl = 0.0f, sr = 0.0f;
#pragma unroll 8
    for (int k = 0; k < GAT_F; ++k) { float v = f[k]; sl += v * a[k]; sr += v * b[k]; }
    el[i] = sl; er[i] = sr;
}

__device__ __forceinline__ float leaky(float v) { return v > 0.0f ? v : NEG_SLOPE * v; }
__device__ __forceinline__ unsigned enc_f32(float f) {
    unsigned u = __float_as_uint(f);
    return (u & 0x80000000u) ? ~u : (u | 0x80000000u);
}

// ---------------- edge pass 1: segment max over dst via monotonic-uint atomicMax ----------------
__global__ void gat_edge_max(const int* __restrict__ src, const int* __restrict__ dst,
                             const float* __restrict__ el, const float* __restrict__ er,
                             unsigned* __restrict__ keys, int neh) {
    int i = blockIdx.x * blockDim.x + threadIdx.x;
    if (i >= neh) return;
    int e = i >> 2, hh = i & 3;
    int s = src[e], d = dst[e];
    float v = leaky(el[s * GAT_H + hh] + er[d * GAT_H + hh]);
    atomicMax(&keys[d * GAT_H + hh], enc_f32(v));
}

// ---------------- decode max keys; empty segment (-inf) -> 0 like reference ----------------
__global__ void gat_node_max(const unsigned* __restrict__ keys, float* __restrict__ emax, int nnh) {
    int i = blockIdx.x * blockDim.x + threadIdx.x;
    if (i >= nnh) return;
    unsigned k = keys[i];
    unsigned u = (k & 0x80000000u) ? (k ^ 0x80000000u) : ~k;
    float m = __uint_as_float(u);
    bool finite = ((u >> 23) & 0xFFu) != 0xFFu;
    emax[i] = finite ? m : 0.0f;
}

// ---------------- edge pass 2: ee = exp(e - emax[dst]); denom[dst] += ee ----------------
__global__ void gat_edge_exp(const int* __restrict__ src, const int* __restrict__ dst,
                             const float* __restrict__ el, const float* __restrict__ er,
                             const float* __restrict__ emax, float* __restrict__ ee,
                             float* __restrict__ denom, int neh) {
    int i = blockIdx.x * blockDim.x + threadIdx.x;
    if (i >= neh) return;
    int e = i >> 2, hh = i & 3;
    int s = src[e], d = dst[e];
    float v = leaky(el[s * GAT_H + hh] + er[d * GAT_H + hh]);
    float x = __expf(v - emax[d * GAT_H + hh]);
    ee[i] = x;
    atomicAdd(&denom[d * GAT_H + hh], x);
}

// ---------------- edge pass 3: rst[dst] += alpha * ft[src] (64 threads/edge, float4 each) ----------------
__global__ void gat_edge_agg(const int* __restrict__ src, const int* __restrict__ dst,
                             const float* __restrict__ ft, const float* __restrict__ ee,
                             const float* __restrict__ denom, float* __restrict__ rst,
                             int total) {
    int t = blockIdx.x * blockDim.x + threadIdx.x;
    if (t >= total) return;
    int e = t >> 6;
    int j = (t & 63) << 2;       // element offset 0..252 within [H*F]
    int hh = j >> 6;             // head for these 4 elements
    int s = src[e], d = dst[e];
    float den = denom[d * GAT_H + hh];
    den = den > 0.0f ? den : 1.0f;
    float a = ee[e * GAT_H + hh] / den;
    const float4 v = *(const float4*)(ft + (size_t)s * 256 + j);
    float* o = rst + (size_t)d * 256 + j;
    atomicAdd(o + 0, a * v.x);
    atomicAdd(o + 1, a * v.y);
    atomicAdd(o + 2, a * v.z);
    atomicAdd(o + 3, a * v.w);
}

// ---------------- head sum + bias: out[n,f] = sum_h (rst[n,h,f] + bias[h,f]) ----------------
__global__ void gat_head_sum(const float* __restrict__ rst, const float* __restrict__ bias,
                             float* __restrict__ out, int nf) {
    int i = blockIdx.x * blockDim.x + threadIdx.x;
    if (i >= nf) return;
    int n = i >> 6, f = i & 63;
    const float* r = rst + (size_t)n * 256 + f;
    float s = r[0] + r[64] + r[128] + r[192];
    s += bias[f] + bias[64 + f] + bias[128 + f] + bias[192 + f];
    out[i] = s;
}

extern "C" void kernel_launch(void* const* d_in, const int* in_sizes, int n_in,
                              void* d_out, int out_size, void* d_ws, size_t ws_size,
                              hipStream_t stream) {
    const float* feat   = (const float*)d_in[0];
    const float* W      = (const float*)d_in[1];
    const float* attn_l = (const float*)d_in[2];
    const float* attn_r = (const float*)d_in[3];
    const float* bias   = (const float*)d_in[4];
    const int*   src    = (const int*)d_in[5];
    const int*   dst    = (const int*)d_in[6];
    float* out = (float*)d_out;

    const int N = in_sizes[0] / GAT_F;   // 50000
    const int E = in_sizes[5];           // 500000
    const int NH = N * GAT_H;
    const int EH = E * GAT_H;
    const int NRST = N * GAT_H * GAT_F;  // 12.8M
    const int NF = N * GAT_F;

    // ---- workspace carve (all float-aligned) ----
    char* p = (char*)d_ws;
    float*    ft    = (float*)p;    p += (size_t)NRST * 4;
    float*    rst   = (float*)p;    p += (size_t)NRST * 4;
    float*    el    = (float*)p;    p += (size_t)NH * 4;
    float*    er    = (float*)p;    p += (size_t)NH * 4;
    float*    emax  = (float*)p;    p += (size_t)NH * 4;
    unsigned* keys  = (unsigned*)p; p += (size_t)NH * 4;
    float*    denom = (float*)p;    p += (size_t)NH * 4;
    float*    ee    = (float*)p;    p += (size_t)EH * 4;
    float*    hA    = (float*)p;    p += (size_t)NF * 4;
    float*    hB    = (float*)p;    p += (size_t)NF * 4;

    const int BLK = 256;
    const int gNH  = (NH + BLK - 1) / BLK;
    const int gEH  = (EH + BLK - 1) / BLK;
    const int gNF  = (NF + BLK - 1) / BLK;
    const int gAGG = (E * 64 + BLK - 1) / BLK;
    const int mtiles = (N + 15) / 16;
    const int waves  = mtiles * 16;          // 16 n-tiles each
    const int gGEMM  = (waves + 3) / 4;      // 4 waves per 128-thread block

    const float* hin = feat;
    for (int l = 0; l < GAT_L; ++l) {
        const float* Wl = W      + (size_t)l * GAT_F * GAT_H * GAT_F;
        const float* al = attn_l + (size_t)l * GAT_H * GAT_F;
        const float* ar = attn_r + (size_t)l * GAT_H * GAT_F;
        const float* bl = bias   + (size_t)l * GAT_H * GAT_F;
        float* hout = (l == GAT_L - 1) ? out : ((l & 1) ? hB : hA);

        gat_init<<<2048, BLK, 0, stream>>>(rst, denom, keys, NRST, NH);
        gat_gemm_wmma<<<gGEMM, 128, 0, stream>>>(hin, Wl, ft, N);
        gat_attn_scores<<<gNH, BLK, 0, stream>>>(ft, al, ar, el, er, NH);
        gat_edge_max<<<gEH, BLK, 0, stream>>>(src, dst, el, er, keys, EH);
        gat_node_max<<<gNH, BLK, 0, stream>>>(keys, emax, NH);
        gat_edge_exp<<<gEH, BLK, 0, stream>>>(src, dst, el, er, emax, ee, denom, EH);
        gat_edge_agg<<<gAGG, BLK, 0, stream>>>(src, dst, ft, ee, denom, rst, E * 64);
        gat_head_sum<<<gNF, BLK, 0, stream>>>(rst, bl, hout, NF);

        hin = hout;
    }
}